// SparsemaxBisect_1580547968598
// MI455X (gfx1250) — compile-verified
//
#include <hip/hip_runtime.h>

// Sparsemax (alpha=2) via 50-iter bisection, rows of length 32000.
// One workgroup per row; row held LDS-resident (loaded via CDNA5 async
// global->LDS b128 ops); bisection runs on a compacted candidate set
// (x > max-1), since relu(x - tau) == 0 for all other elements for every
// tau probed by the bisection (tau >= max-1 always).

#define SPX_D     32000
#define SPX_C4    8000      // float4 chunks per row
#define SPX_BLOCK 1024      // 32 waves (wave32)
#define SPX_NWAVE 32
#define SPX_CAP   2048
#define SPX_NITER 50

struct __align__(16) SpxSmem {
  float row[SPX_D];       // 128000 B, LDS-resident row
  float red[SPX_NWAVE];   // float reduction scratch
  int   ired[SPX_NWAVE];  // int scan scratch
  float bcast[2];         // tau_m, sum_p broadcast
  float cand[SPX_CAP];    // compacted candidates
};

#if defined(__gfx1250__)
#define SPX_ASYNC 1
#else
#define SPX_ASYNC 0
#endif

#if SPX_ASYNC
// Async DMA of 16 bytes/lane from global to LDS (tracked by ASYNCcnt).
// VDST operand = per-lane LDS byte address (low 32 bits of generic pointer
// == wave-relative LDS offset per the aperture mapping), VADDR = u64 global
// address in a VGPR pair.
__device__ __forceinline__ void spx_async_copy16(void* lds_dst, const void* gsrc) {
  unsigned int lds_a = (unsigned int)(unsigned long long)lds_dst;
  unsigned long long g_a = (unsigned long long)gsrc;
  asm volatile("global_load_async_to_lds_b128 %0, %1, off"
               :
               : "v"(lds_a), "v"(g_a)
               : "memory");
}

__device__ __forceinline__ void spx_wait_async() {
#if __has_builtin(__builtin_amdgcn_s_wait_asynccnt)
  __builtin_amdgcn_s_wait_asynccnt(0);
#else
  asm volatile("s_wait_asynccnt 0" ::: "memory");
#endif
}
#endif  // SPX_ASYNC

__device__ __forceinline__ float spx_wave_sum(float v) {
#pragma unroll
  for (int off = 16; off > 0; off >>= 1) v += __shfl_xor(v, off, 32);
  return v;
}

__device__ __forceinline__ float spx_wave_max(float v) {
#pragma unroll
  for (int off = 16; off > 0; off >>= 1) v = fmaxf(v, __shfl_xor(v, off, 32));
  return v;
}

__device__ __forceinline__ float spx_block_sum(float v, float* red, int tid) {
  v = spx_wave_sum(v);
  if ((tid & 31) == 0) red[tid >> 5] = v;
  __syncthreads();
  if (tid < SPX_NWAVE) {
    float r = red[tid];
    r = spx_wave_sum(r);
    if (tid == 0) red[0] = r;
  }
  __syncthreads();
  float out = red[0];
  __syncthreads();
  return out;
}

__device__ __forceinline__ float spx_block_max(float v, float* red, int tid) {
  v = spx_wave_max(v);
  if ((tid & 31) == 0) red[tid >> 5] = v;
  __syncthreads();
  if (tid < SPX_NWAVE) {
    float r = red[tid];
    r = spx_wave_max(r);
    if (tid == 0) red[0] = r;
  }
  __syncthreads();
  float out = red[0];
  __syncthreads();
  return out;
}

// Full-row relu-sum (fallback path only; block-uniform result).
__device__ float spx_fullrow_relusum(SpxSmem* s, int tid, float tau) {
  float acc = 0.0f;
  for (int i = tid; i < SPX_D; i += SPX_BLOCK) acc += fmaxf(s->row[i] - tau, 0.0f);
  return spx_block_sum(acc, s->red, tid);
}

__global__ void __launch_bounds__(SPX_BLOCK)
spx_kernel(const float* __restrict__ X, float* __restrict__ Y) {
  extern __shared__ __align__(16) char spx_raw[];
  SpxSmem* s = reinterpret_cast<SpxSmem*>(spx_raw);
  const int tid = threadIdx.x;
  const size_t rbase = (size_t)blockIdx.x * SPX_D;
  const float4* __restrict__ src4 = reinterpret_cast<const float4*>(X + rbase);
  float4* row4 = reinterpret_cast<float4*>(s->row);

  // ---- 1. load row into LDS (CDNA5 async global->LDS) ----
#pragma unroll
  for (int k = 0; k < 8; ++k) {
    int c = tid + k * SPX_BLOCK;
    if (c < SPX_C4) {
#if SPX_ASYNC
      spx_async_copy16(row4 + c, src4 + c);
#else
      row4[c] = src4[c];
#endif
    }
  }
#if SPX_ASYNC
  spx_wait_async();
#endif
  __syncthreads();

  // ---- 2. row max ----
  float m = -3.402823466e38f;
#pragma unroll
  for (int k = 0; k < 8; ++k) {
    int c = tid + k * SPX_BLOCK;
    if (c < SPX_C4) {
      float4 v = row4[c];
      m = fmaxf(m, fmaxf(fmaxf(v.x, v.y), fmaxf(v.z, v.w)));
    }
  }
  const float maxv = spx_block_max(m, s->red, tid);
  const float thr = maxv - 1.0f;  // tau_lo initial; relu support stays within {x > thr}

  // ---- 3. deterministic candidate compaction (prefix-sum, no atomics) ----
  int cnt_t = 0;
#pragma unroll
  for (int k = 0; k < 8; ++k) {
    int c = tid + k * SPX_BLOCK;
    if (c < SPX_C4) {
      float4 v = row4[c];
      cnt_t += (v.x > thr) + (v.y > thr) + (v.z > thr) + (v.w > thr);
    }
  }
  const int lane = tid & 31;
  const int wave = tid >> 5;
  int incl = cnt_t;
#pragma unroll
  for (int off = 1; off < 32; off <<= 1) {
    int t = __shfl_up(incl, off, 32);
    if (lane >= off) incl += t;
  }
  if (lane == 31) s->ired[wave] = incl;
  __syncthreads();
  if (tid < SPX_NWAVE) {
    int v = s->ired[tid];
#pragma unroll
    for (int off = 1; off < 32; off <<= 1) {
      int t = __shfl_up(v, off, 32);
      if (tid >= off) v += t;
    }
    s->ired[tid] = v;
  }
  __syncthreads();
  const int ncand = s->ired[SPX_NWAVE - 1];
  int idx = (wave ? s->ired[wave - 1] : 0) + (incl - cnt_t);
  if (cnt_t > 0) {
#pragma unroll
    for (int k = 0; k < 8; ++k) {
      int c = tid + k * SPX_BLOCK;
      if (c < SPX_C4) {
        float4 v = row4[c];
        float e0 = v.x, e1 = v.y, e2 = v.z, e3 = v.w;
        if (e0 > thr) { if (idx < SPX_CAP) s->cand[idx] = e0; ++idx; }
        if (e1 > thr) { if (idx < SPX_CAP) s->cand[idx] = e1; ++idx; }
        if (e2 > thr) { if (idx < SPX_CAP) s->cand[idx] = e2; ++idx; }
        if (e3 > thr) { if (idx < SPX_CAP) s->cand[idx] = e3; ++idx; }
      }
    }
  }
  __syncthreads();

  // ---- 4. bisection (matches reference: 50 iters, mask = f_m*f_lo >= 0) ----
  float tau_m, sum_p;
  if (ncand <= SPX_CAP) {
    // Fast path: wave 0 only, shuffle reductions, no barriers in the loop.
    if (tid < 32) {
      float tau_lo = thr;
      float dm = (maxv - (1.0f / (float)SPX_D)) - tau_lo;
      float a = 0.0f;
      for (int i = tid; i < ncand; i += 32) a += fmaxf(s->cand[i] - tau_lo, 0.0f);
      const float f_lo = spx_wave_sum(a) - 1.0f;
      float tm = tau_lo;
      for (int it = 0; it < SPX_NITER; ++it) {
        dm *= 0.5f;
        tm = tau_lo + dm;
        float fm = 0.0f;
        for (int i = tid; i < ncand; i += 32) fm += fmaxf(s->cand[i] - tm, 0.0f);
        fm = spx_wave_sum(fm) - 1.0f;
        if (fm * f_lo >= 0.0f) tau_lo = tm;
      }
      float sp = 0.0f;
      for (int i = tid; i < ncand; i += 32) sp += fmaxf(s->cand[i] - tm, 0.0f);
      sp = spx_wave_sum(sp);
      if (tid == 0) { s->bcast[0] = tm; s->bcast[1] = sp; }
    }
    __syncthreads();
    tau_m = s->bcast[0];
    sum_p = s->bcast[1];
  } else {
    // Fallback: full-row sums with block reductions (block-uniform branch).
    float tau_lo = thr;
    float dm = (maxv - (1.0f / (float)SPX_D)) - tau_lo;
    const float f_lo = spx_fullrow_relusum(s, tid, tau_lo) - 1.0f;
    float tm = tau_lo;
    for (int it = 0; it < SPX_NITER; ++it) {
      dm *= 0.5f;
      tm = tau_lo + dm;
      float fm = spx_fullrow_relusum(s, tid, tm) - 1.0f;
      if (fm * f_lo >= 0.0f) tau_lo = tm;
    }
    tau_m = tm;
    sum_p = spx_fullrow_relusum(s, tid, tm);
  }

  // ---- 5. write p = relu(x - tau_m) / sum_p ----
  float4* dst4 = reinterpret_cast<float4*>(Y + rbase);
#pragma unroll
  for (int k = 0; k < 8; ++k) {
    int c = tid + k * SPX_BLOCK;
    if (c < SPX_C4) {
      float4 v = row4[c];
      float4 p;
      p.x = fmaxf(v.x - tau_m, 0.0f) / sum_p;
      p.y = fmaxf(v.y - tau_m, 0.0f) / sum_p;
      p.z = fmaxf(v.z - tau_m, 0.0f) / sum_p;
      p.w = fmaxf(v.w - tau_m, 0.0f) / sum_p;
      dst4[c] = p;
    }
  }
}

extern "C" void kernel_launch(void* const* d_in, const int* in_sizes, int n_in,
                              void* d_out, int out_size, void* d_ws, size_t ws_size,
                              hipStream_t stream) {
  (void)n_in; (void)out_size; (void)d_ws; (void)ws_size;
  const float* X = (const float*)d_in[0];
  float* Y = (float*)d_out;
  const int rows = in_sizes[0] / SPX_D;
  const size_t shmem = sizeof(SpxSmem);  // ~136 KB of the 320 KB/WGP LDS
  (void)hipFuncSetAttribute(reinterpret_cast<const void*>(spx_kernel),
                            hipFuncAttributeMaxDynamicSharedMemorySize,
                            (int)shmem);
  spx_kernel<<<rows, SPX_BLOCK, shmem, stream>>>(X, Y);
}